// SSIM_68702296867503
// MI455X (gfx1250) — compile-verified
//
#include <hip/hip_runtime.h>
#include <hip/hip_bf16.h>

// SSIM for (16,3,512,512) fp32 via V_WMMA_F32_16X16X4_F32 banded-matmul convolution.
// Horizontal pass fused across the 5 conv fields; band weights precomputed in registers
// from a zero-padded LDS table (no exec-masked gather branches).

typedef float v2f __attribute__((ext_vector_type(2)));
typedef float v8f __attribute__((ext_vector_type(8)));

#define SSIM_C1 6.5025f    // (0.01*255)^2
#define SSIM_C2 58.5225f   // (0.03*255)^2
#define H 512
#define W 512
#define OUTW 502           // 512 - 11 + 1
#define TILES 32           // ceil(502/16)

__global__ void ssim_zero_ws(float* ws) {
    if (threadIdx.x < 16) ws[threadIdx.x] = 0.0f;
}

__global__ void ssim_finalize(const float* __restrict__ ws, float* __restrict__ out) {
    int b = threadIdx.x;
    if (b < 16) out[b] = ws[b] * (1.0f / (3.0f * (float)OUTW * (float)OUTW));
}

__device__ __forceinline__ float fieldVal(int f, float x, float y) {
    switch (f) {
        case 0:  return x;
        case 1:  return y;
        case 2:  return x * x;
        case 3:  return y * y;
        default: return x * y;
    }
}

__global__ __launch_bounds__(32)
void ssim_tile_kernel(const float* __restrict__ raw, const float* __restrict__ dst,
                      float* __restrict__ ws) {
    // One wave per 16x16 output tile.
    const int lane = threadIdx.x;          // 0..31
    const int half = lane >> 4;            // 0: lanes 0-15, 1: lanes 16-31
    const int lm   = lane & 15;

    const int tile = blockIdx.x;           // 0..1023
    const int ty = tile >> 5;              // tile row
    const int tx = tile & 31;              // tile col
    const int ch = blockIdx.y;             // 0..2
    const int b  = blockIdx.z;             // 0..15

    const int oy = ty * 16;                // output tile origin = input patch origin
    const int ox = tx * 16;

    const size_t imgBase = ((size_t)(b * 3 + ch)) * H * W;

    __shared__ float xs[26 * 28];          // raw patch *255, cols 26..27 zero pad
    __shared__ float ys[26 * 28];          // dst patch *255
    __shared__ float mid[5][28 * 16];      // per-field horizontal intermediates (rows 26,27 zero)
    __shared__ float gpad[44];             // zero-padded Gaussian band: gpad[d+15] = g[d] or 0

    // CDNA5 prefetch hint (global_prefetch_b8) on the patch base.
    __builtin_prefetch(raw + imgBase + (size_t)oy * W + ox, 0, 1);
    __builtin_prefetch(dst + imgBase + (size_t)oy * W + ox, 0, 1);

    // Gaussian window (sigma = 1.5), normalized; zero-padded so band lookups need no branch.
    {
        float s = 0.0f;
        #pragma unroll
        for (int i = 0; i < 11; ++i) {
            float d = (float)i - 5.0f;
            s += expf(-(d * d) / 4.5f);
        }
        #pragma unroll
        for (int base = 0; base < 44; base += 32) {
            int i = base + lane;
            if (i < 44) {
                float d = (float)(i - 20);                 // tap index (i-15) - 5
                float g = expf(-(d * d) / 4.5f) / s;
                gpad[i] = (i >= 15 && i < 26) ? g : 0.0f;  // cndmask, no branch around DS
            }
        }
    }

    // Stage 26x26 patch into LDS (stride 28; pad columns zeroed so K-padding is exact 0).
    for (int idx = lane; idx < 26 * 28; idx += 32) {
        int r = idx / 28;
        int c = idx - r * 28;
        float xv = 0.0f, yv = 0.0f;
        if (c < 26) {
            int gr = min(oy + r, H - 1);
            int gc = min(ox + c, W - 1);
            size_t off = imgBase + (size_t)gr * W + gc;
            xv = raw[off] * 255.0f;
            yv = dst[off] * 255.0f;
        }
        xs[idx] = xv;
        ys[idx] = yv;
    }
    // Zero the two K-padding rows (26,27) of every mid plane; chains only write rows 0..25.
    #pragma unroll
    for (int p = 0; p < 5; ++p)
        mid[p][(26 + half) * 16 + lm] = 0.0f;
    __syncthreads();

    // Band-weight fragments, shared by horizontal-B and vertical-A (same generator):
    // wf[q].{x,y} = W[kb][lm], W[kb+1][lm] with W[k][n] = g[k-n] (0 outside band).
    v2f wf[7];
    {
        const int base = half * 2 - lm + 15;   // gpad index for (kb=q*4+half*2, n=lm)
        #pragma unroll
        for (int q = 0; q < 7; ++q) {
            wf[q].x = gpad[base + q * 4];
            wf[q].y = gpad[base + q * 4 + 1];
        }
    }

    // ---- horizontal pass, fused over the 5 fields: Mid[f] = In(f) x BandW ----
    v8f h[5][2];
    #pragma unroll
    for (int f = 0; f < 5; ++f) {
        h[f][0] = (v8f){0.f,0.f,0.f,0.f,0.f,0.f,0.f,0.f};
        h[f][1] = (v8f){0.f,0.f,0.f,0.f,0.f,0.f,0.f,0.f};
    }
    #pragma unroll
    for (int q = 0; q < 7; ++q) {
        const int kb = q * 4 + half * 2;
        const v2f bw = wf[q];
        const int r0 = lm * 28 + kb;           // chain 0: patch rows 0..15
        const int r1 = (10 + lm) * 28 + kb;    // chain 1: patch rows 10..25
        const float x00 = xs[r0], x01 = xs[r0 + 1];
        const float y00 = ys[r0], y01 = ys[r0 + 1];
        const float x10 = xs[r1], x11 = xs[r1 + 1];
        const float y10 = ys[r1], y11 = ys[r1 + 1];
        #pragma unroll
        for (int f = 0; f < 5; ++f) {
            v2f a0; a0.x = fieldVal(f, x00, y00); a0.y = fieldVal(f, x01, y01);
            h[f][0] = __builtin_amdgcn_wmma_f32_16x16x4_f32(false, a0, false, bw,
                                                            (short)0, h[f][0], false, false);
            v2f a1; a1.x = fieldVal(f, x10, y10); a1.y = fieldVal(f, x11, y11);
            h[f][1] = __builtin_amdgcn_wmma_f32_16x16x4_f32(false, a1, false, bw,
                                                            (short)0, h[f][1], false, false);
        }
    }

    // Scatter D fragments to mid planes (D layout: VGPR i -> M = i + half*8).
    #pragma unroll
    for (int f = 0; f < 5; ++f) {
        #pragma unroll
        for (int i = 0; i < 8; ++i) {
            mid[f][(i + half * 8) * 16 + lm]      = h[f][0][i];
            mid[f][(10 + i + half * 8) * 16 + lm] = h[f][1][i];
        }
    }
    __syncthreads();

    // ---- vertical pass per field: Out[f] = BandW(16x28) x Mid[f](28x16) ----
    v8f outAcc[5];
    #pragma unroll
    for (int f = 0; f < 5; ++f) {
        v8f acc = (v8f){0.f,0.f,0.f,0.f,0.f,0.f,0.f,0.f};
        #pragma unroll
        for (int q = 0; q < 7; ++q) {
            const int kb = q * 4 + half * 2;
            v2f bm;
            bm.x = mid[f][kb * 16 + lm];
            bm.y = mid[f][(kb + 1) * 16 + lm];
            acc = __builtin_amdgcn_wmma_f32_16x16x4_f32(false, wf[q], false, bm,
                                                        (short)0, acc, false, false);
        }
        outAcc[f] = acc;
    }

    // ---- SSIM map + masked tile reduction ----
    float part = 0.0f;
    #pragma unroll
    for (int i = 0; i < 8; ++i) {
        const int orow = oy + i + half * 8;
        const int ocol = ox + lm;
        const float ux = outAcc[0][i];
        const float uy = outAcc[1][i];
        const float ux2 = ux * ux;
        const float uy2 = uy * uy;
        const float uxy = ux * uy;
        const float vx  = outAcc[2][i] - ux2;
        const float vy  = outAcc[3][i] - uy2;
        const float cxy = outAcc[4][i] - uxy;
        const float numer = (2.0f * uxy + SSIM_C1) * (2.0f * cxy + SSIM_C2);
        const float denom = (ux2 + uy2 + SSIM_C1) * (vx + vy + SSIM_C2);
        const float sv = numer / denom;
        part += (orow < OUTW && ocol < OUTW) ? sv : 0.0f;
    }

    // wave32 reduction
    #pragma unroll
    for (int off = 16; off > 0; off >>= 1)
        part += __shfl_down(part, off, 32);

    if (lane == 0)
        atomicAdd(&ws[b], part);
}

extern "C" void kernel_launch(void* const* d_in, const int* in_sizes, int n_in,
                              void* d_out, int out_size, void* d_ws, size_t ws_size,
                              hipStream_t stream) {
    const float* raw = (const float*)d_in[0];
    const float* dst = (const float*)d_in[1];
    float* out = (float*)d_out;
    float* ws  = (float*)d_ws;

    ssim_zero_ws<<<1, 32, 0, stream>>>(ws);
    dim3 grid(TILES * TILES, 3, 16);
    ssim_tile_kernel<<<grid, 32, 0, stream>>>(raw, dst, ws);
    ssim_finalize<<<1, 32, 0, stream>>>(ws, out);
}